// GCN_3453153706308
// MI455X (gfx1250) — compile-verified
//
#include <hip/hip_runtime.h>
#include <hip/hip_bf16.h>

#define NN  100000
#define EE  3200000
#define FIN 128
#define HH  64
#define CCL 2
#define GG  512

typedef __attribute__((ext_vector_type(2))) float v2f;
typedef __attribute__((ext_vector_type(8))) float v8f;

// ---------------------------------------------------------------------------
// Init: deg=1 (self loop), zero aggregation buffers / pools, pmax=-inf
// ---------------------------------------------------------------------------
__global__ void init_kernel(float* __restrict__ agg1, float* __restrict__ agg2,
                            float* __restrict__ deg,  float* __restrict__ psum,
                            float* __restrict__ pmax, float* __restrict__ cnt) {
    int i = blockIdx.x * 256 + threadIdx.x;
    if (i < NN * HH) { agg1[i] = 0.f; agg2[i] = 0.f; }
    if (i < NN)      deg[i] = 1.0f;
    if (i < GG * HH) { psum[i] = 0.f; pmax[i] = -__builtin_inff(); }
    if (i < GG)      cnt[i] = 0.f;
}

// ---------------------------------------------------------------------------
// In-degree over dst (self loop already folded in via deg=1 init)
// ---------------------------------------------------------------------------
__global__ void degree_kernel(const int* __restrict__ ei, float* __restrict__ deg) {
    int e = blockIdx.x * 256 + threadIdx.x;
    if (e < EE) atomicAdd(&deg[ei[EE + e]], 1.0f);
}

__global__ void dinv_kernel(float* __restrict__ deg) {
    int i = blockIdx.x * 256 + threadIdx.x;
    if (i < NN) deg[i] = rsqrtf(fmaxf(deg[i], 1.0f));
}

// ---------------------------------------------------------------------------
// WMMA f32 GEMM: Y[N,64] = X[N,KD] @ W[KD,64]
// One wave32 computes a 16x64 tile via 4 v8f accumulators, K stepped by 4
// with V_WMMA_F32_16X16X4_F32.
// A layout (16x4 f32): lane(0..15)=row, VGPRs = K {0,1}; lanes 16..31 K {2,3}.
// B layout (4x16 f32): lane%16 = col,  VGPRs = K {0,1}; lanes 16..31 K {2,3}.
// C/D layout: lane%16 = col, VGPR v = row v + 8*(lane/16).
// W is staged in LDS K-pair-interleaved: lds[(k/2)*128 + 2*col + (k&1)], so
// each B fragment is a single aligned ds_load_b64 into an even VGPR pair
// (no post-load register shuffles before the WMMA).
// ---------------------------------------------------------------------------
template <int KD>
__global__ __launch_bounds__(256) void gemm_wmma(const float* __restrict__ X,
                                                 const float* __restrict__ W,
                                                 float* __restrict__ Y) {
    __shared__ float ldsW[KD * HH];
    int tid = threadIdx.x;
    // Stage W: read float4 rows, write pair-interleaved layout.
    for (int idx = tid; idx < KD * (HH / 4); idx += 256) {
        int k  = idx >> 4;            // 0..KD-1
        int c4 = (idx & 15) * 4;      // 0,4,...,60
        float4 w = ((const float4*)(W + (size_t)k * HH))[c4 >> 2];
        int base = (k >> 1) * (2 * HH) + 2 * c4 + (k & 1);
        ldsW[base]     = w.x;
        ldsW[base + 2] = w.y;
        ldsW[base + 4] = w.z;
        ldsW[base + 6] = w.w;
    }
    __syncthreads();

    int wave = tid >> 5;
    int lane = tid & 31;
    int tileRow = (blockIdx.x * 8 + wave) * 16;
    if (tileRow >= NN) return;        // wave-uniform: EXEC stays all-1s inside

    int row16 = lane & 15;
    int kh    = lane >> 4;            // 0: K pair {0,1}; 1: K pair {2,3}
    const float* xrow = X + (size_t)(tileRow + row16) * KD;
    int col = lane & 15;
    const v2f* wp = (const v2f*)ldsW; // v2f units: pair p at wp[p*HH + col]

    v8f acc0 = {}, acc1 = {}, acc2 = {}, acc3 = {};

#pragma unroll 4
    for (int k0 = 0; k0 < KD; k0 += 4) {
        int kk = k0 + 2 * kh;
        v2f a = *(const v2f*)(xrow + kk);            // global_load_b64
        const v2f* wrow = wp + (size_t)((k0 >> 1) + kh) * HH;
        v2f b0 = wrow[col];                          // ds_load_b64 each
        v2f b1 = wrow[col + 16];
        v2f b2 = wrow[col + 32];
        v2f b3 = wrow[col + 48];
        acc0 = __builtin_amdgcn_wmma_f32_16x16x4_f32(false, a, false, b0, (short)0, acc0, false, false);
        acc1 = __builtin_amdgcn_wmma_f32_16x16x4_f32(false, a, false, b1, (short)0, acc1, false, false);
        acc2 = __builtin_amdgcn_wmma_f32_16x16x4_f32(false, a, false, b2, (short)0, acc2, false, false);
        acc3 = __builtin_amdgcn_wmma_f32_16x16x4_f32(false, a, false, b3, (short)0, acc3, false, false);
    }

    // NN % 16 == 0, so every row of an in-range tile is in range: no guards.
    int mbase = 8 * kh;
#pragma unroll
    for (int v = 0; v < 8; v++) {
        float* yr = Y + (size_t)(tileRow + mbase + v) * HH + col;
        yr[0]  = acc0[v];
        yr[16] = acc1[v];
        yr[32] = acc2[v];
        yr[48] = acc3[v];
    }
}

// ---------------------------------------------------------------------------
// Edge scatter: AGG[dst] += H[src] * dinv[src]*dinv[dst]
// Half-wave (16 lanes) per edge, float4 per lane -> coalesced 64-float rows.
// ---------------------------------------------------------------------------
__global__ void scatter_kernel(const int* __restrict__ ei, const float* __restrict__ dinv,
                               const float* __restrict__ H, float* __restrict__ AGG) {
    long long gid = (long long)blockIdx.x * 256 + threadIdx.x;
    long long e = gid >> 4;
    if (e >= EE) return;
    int f = (int)(gid & 15) * 4;
    int s = ei[e], d = ei[EE + e];
    float nrm = dinv[s] * dinv[d];
    float4 h = *(const float4*)(H + (size_t)s * HH + f);
    float* o = AGG + (size_t)d * HH + f;
    atomicAdd(o + 0, h.x * nrm);
    atomicAdd(o + 1, h.y * nrm);
    atomicAdd(o + 2, h.z * nrm);
    atomicAdd(o + 3, h.w * nrm);
}

// ---------------------------------------------------------------------------
// Finalize conv: AGG += self-loop (H*dinv^2) + bias, optional ReLU (in place)
// ---------------------------------------------------------------------------
__global__ void finalize_kernel(float* __restrict__ AGG, const float* __restrict__ Hpre,
                                const float* __restrict__ dinv, const float* __restrict__ bias,
                                int relu) {
    int i = blockIdx.x * 256 + threadIdx.x;
    if (i >= NN * HH) return;
    int n = i >> 6, f = i & 63;
    float di = dinv[n];
    float v = AGG[i] + Hpre[i] * di * di + bias[f];
    if (relu) v = fmaxf(v, 0.f);
    AGG[i] = v;
}

// ---------------------------------------------------------------------------
// Pooling: per-graph sum / max / count
// ---------------------------------------------------------------------------
__device__ inline void atomicMaxF(float* addr, float val) {
    if (val >= 0.f) atomicMax((int*)addr, __float_as_int(val));
    else            atomicMin((unsigned int*)addr, __float_as_uint(val));
}

__global__ void pool_kernel(const float* __restrict__ H2, const int* __restrict__ batch,
                            float* __restrict__ psum, float* __restrict__ pmax,
                            float* __restrict__ cnt) {
    int i = blockIdx.x * 256 + threadIdx.x;
    if (i >= NN * HH) return;
    int n = i >> 6, f = i & 63;
    int g = batch[n];
    float v = H2[i];
    atomicAdd(&psum[g * HH + f], v);
    atomicMaxF(&pmax[g * HH + f], v);
    if (f == 0) atomicAdd(&cnt[g], 1.0f);
}

// ---------------------------------------------------------------------------
// MLP head + log_softmax: one 64-thread block per graph (tiny compute)
// ---------------------------------------------------------------------------
__global__ __launch_bounds__(64) void mlp_kernel(const float* __restrict__ psum,
                                                 const float* __restrict__ pmax,
                                                 const float* __restrict__ cnt,
                                                 const float* __restrict__ l1w,
                                                 const float* __restrict__ l1b,
                                                 const float* __restrict__ l2w,
                                                 const float* __restrict__ l2b,
                                                 float* __restrict__ out) {
    __shared__ float z[3 * HH];
    __shared__ float hid[HH];
    int g = blockIdx.x, t = threadIdx.x;
    float c = fmaxf(cnt[g], 1.0f);
    float s = psum[g * HH + t];
    z[t]          = s / c;          // mean
    z[HH + t]     = pmax[g * HH + t];
    z[2 * HH + t] = s;              // sum
    __syncthreads();
    float acc = l1b[t];
#pragma unroll 8
    for (int k = 0; k < 3 * HH; k++) acc += z[k] * l1w[k * HH + t];
    hid[t] = fmaxf(acc, 0.f);
    __syncthreads();
    if (t == 0) {
        float o0 = l2b[0], o1 = l2b[1];
        for (int k = 0; k < HH; k++) { o0 += hid[k] * l2w[k * 2]; o1 += hid[k] * l2w[k * 2 + 1]; }
        float m = fmaxf(o0, o1);
        float lse = m + logf(expf(o0 - m) + expf(o1 - m));
        out[g * 2 + 0] = o0 - lse;
        out[g * 2 + 1] = o1 - lse;
    }
}

// ---------------------------------------------------------------------------
extern "C" void kernel_launch(void* const* d_in, const int* in_sizes, int n_in,
                              void* d_out, int out_size, void* d_ws, size_t ws_size,
                              hipStream_t stream) {
    const float* x     = (const float*)d_in[0];
    const int*   ei    = (const int*)  d_in[1];
    const int*   batch = (const int*)  d_in[2];
    const float* W1    = (const float*)d_in[3];
    const float* b1    = (const float*)d_in[4];
    const float* W2    = (const float*)d_in[5];
    const float* b2    = (const float*)d_in[6];
    const float* l1w   = (const float*)d_in[7];
    const float* l1b   = (const float*)d_in[8];
    const float* l2w   = (const float*)d_in[9];
    const float* l2b   = (const float*)d_in[10];
    float* out = (float*)d_out;

    float* ws   = (float*)d_ws;
    float* bufA = ws;                          // h1 (x@W1), later h2pre
    float* bufB = bufA + (size_t)NN * HH;      // agg1, later h1post (in place)
    float* bufC = bufB + (size_t)NN * HH;      // agg2, later h2 (in place)
    float* deg  = bufC + (size_t)NN * HH;      // deg -> dinv (in place)
    float* psum = deg + NN;
    float* pmax = psum + (size_t)GG * HH;
    float* cnt  = pmax + (size_t)GG * HH;

    const int nb_nh   = (NN * HH + 255) / 256;
    const int nb_e    = (EE + 255) / 256;
    const int nb_n    = (NN + 255) / 256;
    const int nb_gemm = ((NN + 15) / 16 + 7) / 8;                     // 782
    const int nb_scat = (int)(((long long)EE * 16 + 255) / 256);      // 200000

    init_kernel<<<nb_nh, 256, 0, stream>>>(bufB, bufC, deg, psum, pmax, cnt);
    degree_kernel<<<nb_e, 256, 0, stream>>>(ei, deg);
    dinv_kernel<<<nb_n, 256, 0, stream>>>(deg);

    // Layer 1: h1 = x @ W1 ; aggregate ; + self-loop + b1 ; ReLU
    gemm_wmma<FIN><<<nb_gemm, 256, 0, stream>>>(x, W1, bufA);
    scatter_kernel<<<nb_scat, 256, 0, stream>>>(ei, deg, bufA, bufB);
    finalize_kernel<<<nb_nh, 256, 0, stream>>>(bufB, bufA, deg, b1, 1);

    // Layer 2: h2pre = h1post @ W2 ; aggregate ; + self-loop + b2
    gemm_wmma<HH><<<nb_gemm, 256, 0, stream>>>(bufB, W2, bufA);
    scatter_kernel<<<nb_scat, 256, 0, stream>>>(ei, deg, bufA, bufC);
    finalize_kernel<<<nb_nh, 256, 0, stream>>>(bufC, bufA, deg, b2, 0);

    // Pool + MLP head
    pool_kernel<<<nb_nh, 256, 0, stream>>>(bufC, batch, psum, pmax, cnt);
    mlp_kernel<<<GG, 64, 0, stream>>>(psum, pmax, cnt, l1w, l1b, l2w, l2b, out);
}